// GCN_85091892068866
// MI455X (gfx1250) — compile-verified
//
#include <hip/hip_runtime.h>
#include <hip/hip_bf16.h>
#include <climits>

typedef __attribute__((ext_vector_type(2))) float v2f;
typedef __attribute__((ext_vector_type(4))) float v4f;
typedef __attribute__((ext_vector_type(8))) float v8f;

static constexpr float NEG_SLOPE = 0.2f;
static constexpr float EPS_BN = 1e-5f;

// ---------- helpers ----------
__device__ __forceinline__ int f2ord(float f) {
  int i = __float_as_int(f);
  return i >= 0 ? i : i ^ 0x7FFFFFFF;   // monotone float->int map (involution)
}
__device__ __forceinline__ float ord2f(int i) {
  return __int_as_float(i >= 0 ? i : i ^ 0x7FFFFFFF);
}
__device__ __forceinline__ float lrelu(float v) { return v > 0.0f ? v : NEG_SLOPE * v; }

__global__ void fill_f32(float* __restrict__ p, float v, long long n) {
  long long i = (long long)blockIdx.x * blockDim.x + threadIdx.x;
  if (i < n) p[i] = v;
}
__global__ void fill_i32(int* __restrict__ p, int v, long long n) {
  long long i = (long long)blockIdx.x * blockDim.x + threadIdx.x;
  if (i < n) p[i] = v;
}

// ---------- dense GEMM on the matrix pipe: D[M,NCOL] = A[M,K] * B[K,NCOL] ----------
// f32 WMMA 16x16x4. 8 waves / 256-thread block, each wave owns a 16-row tile and
// NCOL/16 accumulators.
//
// A 16x4 frag (ISA 7.12.2): lanes 0-15 hold M=lane,{K0,K1}; lanes 16-31 M=lane-16,{K2,K3}.
// B 4x16 mirrors it with N=lane&15. C/D: VGPR v -> row v (lanes 0-15) / v+8 (lanes 16-31).
//
// B is staged in LDS with K-row PAIRS interleaved:
//   Bs2[(k/2)*(2*NCOL) + 2*col + (k&1)] = B[k][col]
// so each lane's B fragment {B[kp][col], B[kp+1][col]} is ONE aligned ds_load_b64,
// and consecutive lanes are 8B apart (conflict-free).
//
// Out-of-range rows: a row of A only affects the same row of D, and stores are
// guarded, so the A address is merely CLAMPED -> no divergent branches in the
// hot loop (WMMA requires full-EXEC anyway).
template<int K, int NCOL>
__global__ __launch_bounds__(256) void gemm_wmma(const float* __restrict__ A,
                                                 const float* __restrict__ B,
                                                 float* __restrict__ D, int M) {
  constexpr int NT = NCOL / 16;
  __shared__ float Bs2[K * NCOL];
  for (int i = threadIdx.x; i < K * NCOL; i += 256) {
    int k = i / NCOL, col = i % NCOL;
    Bs2[(k >> 1) * (2 * NCOL) + col * 2 + (k & 1)] = B[i];
  }
  __syncthreads();

  const int wave  = threadIdx.x >> 5;
  const int lane  = threadIdx.x & 31;
  const int row0  = (blockIdx.x * 8 + wave) * 16;
  const int half  = lane >> 4;
  const int l16   = lane & 15;

  int arow = row0 + l16;
  if (arow >= M) arow = M - 1;                       // clamp, don't predicate
  const float* ap = A + (size_t)arow * K + 2 * half; // lane's {k+kbase, k+kbase+1}
  const float* bbase = Bs2 + (size_t)half * (2 * NCOL) + l16 * 2;

  v8f acc[NT] = {};
#pragma unroll
  for (int k = 0; k < K; k += 4) {
    v2f a = *(const v2f*)(ap + k);                   // 8B aligned
    const float* brow = bbase + (size_t)(k / 2) * (2 * NCOL);
#pragma unroll
    for (int t = 0; t < NT; ++t) {
      v2f b = *(const v2f*)(brow + t * 32);          // one ds_load_b64
      acc[t] = __builtin_amdgcn_wmma_f32_16x16x4_f32(
          /*neg_a=*/false, a, /*neg_b=*/false, b,
          /*c_mod=*/(short)0, acc[t], /*reuse_a=*/false, /*reuse_b=*/false);
    }
  }

  float* drow = D + (size_t)(row0 + 8 * half) * NCOL + l16;
  if (row0 + 16 <= M) {                              // uniform fast path
#pragma unroll
    for (int t = 0; t < NT; ++t)
#pragma unroll
      for (int v = 0; v < 8; ++v) drow[(size_t)v * NCOL + t * 16] = acc[t][v];
  } else {
#pragma unroll
    for (int t = 0; t < NT; ++t)
#pragma unroll
      for (int v = 0; v < 8; ++v) {
        int r = row0 + 8 * half + v;
        if (r < M) D[(size_t)r * NCOL + t * 16 + l16] = acc[t][v];
      }
  }
}

// ---------- per-node attention scalars: a_s[n,h] = <h[n,h,:], att_src[h,:]> ----------
template<int C>
__global__ void att_scores(const float* __restrict__ h, const float* __restrict__ asrc,
                           const float* __restrict__ adst, float* __restrict__ a_s,
                           float* __restrict__ a_d, int nNodes) {
  int i = blockIdx.x * blockDim.x + threadIdx.x;
  if (i >= nNodes * 8) return;
  int n = i >> 3, hh = i & 7;
  const float* hp = h + (size_t)n * (8 * C) + hh * C;
  float s = 0.f, d = 0.f;
#pragma unroll
  for (int c = 0; c < C; ++c) {
    float v = hp[c];
    s += v * asrc[hh * C + c];
    d += v * adst[hh * C + c];
  }
  a_s[i] = s;
  a_d[i] = d;
}

__device__ __forceinline__ void load8(const float* __restrict__ p, float* out) {
  v4f lo = *(const v4f*)(p);       // 32B-aligned region -> two b128 loads
  v4f hi = *(const v4f*)(p + 4);
#pragma unroll
  for (int j = 0; j < 4; ++j) { out[j] = lo[j]; out[4 + j] = hi[j]; }
}

// ---------- segment max over incoming edges (ordered-int atomicMax) ----------
__global__ void edge_max(const float* __restrict__ a_s, const float* __restrict__ a_d,
                         const int* __restrict__ ei, int* __restrict__ m_ord,
                         int nE, int nNodes) {
  int e = blockIdx.x * blockDim.x + threadIdx.x;
  if (e >= nE + nNodes) return;
  int src, dst;
  if (e < nE) { src = ei[e]; dst = ei[nE + e]; }
  else        { src = dst = e - nE; }
  float as[8], ad[8];
  load8(a_s + (size_t)src * 8, as);
  load8(a_d + (size_t)dst * 8, ad);
#pragma unroll
  for (int h = 0; h < 8; ++h)
    atomicMax(&m_ord[(size_t)dst * 8 + h], f2ord(lrelu(as[h] + ad[h])));
}

// ---------- ex = exp(e - m[dst]); s[dst] += ex ----------
__global__ void edge_exp_sum(const float* __restrict__ a_s, const float* __restrict__ a_d,
                             const int* __restrict__ ei, const int* __restrict__ m_ord,
                             float* __restrict__ ex, float* __restrict__ s,
                             int nE, int nNodes) {
  int e = blockIdx.x * blockDim.x + threadIdx.x;
  if (e >= nE + nNodes) return;
  int src, dst;
  if (e < nE) { src = ei[e]; dst = ei[nE + e]; }
  else        { src = dst = e - nE; }
  float as[8], ad[8];
  load8(a_s + (size_t)src * 8, as);
  load8(a_d + (size_t)dst * 8, ad);
  v4f t0, t1;
#pragma unroll
  for (int h = 0; h < 8; ++h) {
    float v = lrelu(as[h] + ad[h]);
    float t = __expf(v - ord2f(m_ord[(size_t)dst * 8 + h]));
    if (h < 4) t0[h] = t; else t1[h - 4] = t;
    unsafeAtomicAdd(&s[(size_t)dst * 8 + h], t);
  }
  *(v4f*)(ex + (size_t)e * 8)     = t0;   // 32B-aligned: two b128 stores
  *(v4f*)(ex + (size_t)e * 8 + 4) = t1;
}

// ---------- out[dst,h,:] += h[src,h,:] * (ex / s[dst,h]) ----------
template<int C>
__global__ void aggregate(const float* __restrict__ hfeat, const float* __restrict__ ex,
                          const float* __restrict__ s, const int* __restrict__ ei,
                          float* __restrict__ out, int nE, int nNodes) {
  long long idx = (long long)blockIdx.x * blockDim.x + threadIdx.x;
  long long total = (long long)(nE + nNodes) * 8;
  if (idx >= total) return;
  int e = (int)(idx >> 3);
  int h = (int)(idx & 7);
  int src, dst;
  if (e < nE) { src = ei[e]; dst = ei[nE + e]; }
  else        { src = dst = e - nE; }
  float alpha = ex[(size_t)e * 8 + h] / s[(size_t)dst * 8 + h];
  const float* hp = hfeat + (size_t)src * (8 * C) + h * C;   // 16B aligned
  float* op = out + (size_t)dst * (8 * C) + h * C;
  float hv[C];
#pragma unroll
  for (int q = 0; q < C / 4; ++q) {
    v4f v = *(const v4f*)(hp + q * 4);
#pragma unroll
    for (int j = 0; j < 4; ++j) hv[q * 4 + j] = v[j];
  }
#pragma unroll
  for (int c = 0; c < C; ++c) unsafeAtomicAdd(&op[c], hv[c] * alpha);
}

// ---------- BatchNorm stats (32 channels): LDS ds_add_f32 reduction + global add ----------
__global__ __launch_bounds__(256) void bn_stats(const float* __restrict__ h,
                                                float* __restrict__ stats, int n) {
  __shared__ float ssum[32];
  __shared__ float ssq[32];
  if (threadIdx.x < 32) { ssum[threadIdx.x] = 0.f; ssq[threadIdx.x] = 0.f; }
  __syncthreads();
  int i = blockIdx.x * blockDim.x + threadIdx.x;
  if (i < n) {
    const float* hp = h + (size_t)i * 32;
#pragma unroll
    for (int c = 0; c < 32; ++c) {
      float v = hp[c];
      atomicAdd(&ssum[c], v);        // ds_add_f32
      atomicAdd(&ssq[c],  v * v);
    }
  }
  __syncthreads();
  if (threadIdx.x < 32) {
    unsafeAtomicAdd(&stats[threadIdx.x],      ssum[threadIdx.x]);
    unsafeAtomicAdd(&stats[32 + threadIdx.x], ssq[threadIdx.x]);
  }
}

// ---------- (h-mu)*rsqrt(var+eps)*gamma+beta, ReLU; in place ----------
__global__ void bn_apply(float* __restrict__ h, const float* __restrict__ stats,
                         const float* __restrict__ gamma, const float* __restrict__ beta,
                         int n) {
  long long i = (long long)blockIdx.x * blockDim.x + threadIdx.x;
  if (i >= (long long)n * 32) return;
  int c = (int)(i & 31);
  float inv_n = 1.0f / (float)n;
  float mu  = stats[c] * inv_n;
  float var = stats[32 + c] * inv_n - mu * mu;   // biased variance
  float y = (h[i] - mu) * rsqrtf(var + EPS_BN) * gamma[c] + beta[c];
  h[i] = y > 0.f ? y : 0.f;
}

// ---------- mean over heads + bias2 ----------
__global__ void finalize(const float* __restrict__ out2, const float* __restrict__ bias2,
                         float* __restrict__ out, int n) {
  long long i = (long long)blockIdx.x * blockDim.x + threadIdx.x;
  if (i >= (long long)n * 8) return;
  int nd = (int)(i >> 3);
  int c  = (int)(i & 7);
  const float* p = out2 + (size_t)nd * 64 + c;
  float s = 0.f;
#pragma unroll
  for (int h = 0; h < 8; ++h) s += p[h * 8];
  out[i] = 0.125f * s + bias2[c];
}

// ---------- host ----------
extern "C" void kernel_launch(void* const* d_in, const int* in_sizes, int n_in,
                              void* d_out, int out_size, void* d_ws, size_t ws_size,
                              hipStream_t stream) {
  const float* x        = (const float*)d_in[0];
  const int*   ei       = (const int*)d_in[1];
  const float* W1       = (const float*)d_in[2];
  const float* att_src1 = (const float*)d_in[3];
  const float* att_dst1 = (const float*)d_in[4];
  // d_in[5] = bias1: a per-channel additive constant cancels exactly under BatchNorm.
  const float* gamma    = (const float*)d_in[6];
  const float* beta     = (const float*)d_in[7];
  const float* W2       = (const float*)d_in[8];
  const float* att_src2 = (const float*)d_in[9];
  const float* att_dst2 = (const float*)d_in[10];
  const float* bias2    = (const float*)d_in[11];
  float* out = (float*)d_out;

  const int N  = in_sizes[0] / 128;   // 100000
  const int E  = in_sizes[1] / 2;     // 1600000
  const int Ep = E + N;

  // workspace carve-up (256B aligned)
  char* ws = (char*)d_ws;
  size_t off = 0;
  auto carve = [&](size_t bytes) {
    void* p = ws + off;
    off = (off + bytes + 255) & ~(size_t)255;
    return p;
  };
  float* h1    = (float*)carve((size_t)N * 32 * 4);
  float* out1  = (float*)carve((size_t)N * 32 * 4);
  float* h2    = (float*)carve((size_t)N * 64 * 4);
  float* out2  = (float*)carve((size_t)N * 64 * 4);
  float* a_s   = (float*)carve((size_t)N * 8 * 4);
  float* a_d   = (float*)carve((size_t)N * 8 * 4);
  int*   m_ord = (int*)  carve((size_t)N * 8 * 4);
  float* ssum  = (float*)carve((size_t)N * 8 * 4);
  float* ex    = (float*)carve((size_t)Ep * 8 * 4);
  float* stats = (float*)carve(64 * 4);
  (void)ws_size; (void)n_in; (void)out_size;

  auto cdiv = [](long long a, long long b) { return (unsigned)((a + b - 1) / b); };
  const int T = 256;
  const unsigned gemm_blocks = cdiv(((long long)N + 15) / 16, 8);

  // ---- layer 1 ----
  fill_f32<<<cdiv((long long)N * 32, T), T, 0, stream>>>(out1, 0.f, (long long)N * 32);
  fill_f32<<<cdiv((long long)N * 8,  T), T, 0, stream>>>(ssum, 0.f, (long long)N * 8);
  fill_i32<<<cdiv((long long)N * 8,  T), T, 0, stream>>>(m_ord, INT_MIN, (long long)N * 8);
  fill_f32<<<1, 64, 0, stream>>>(stats, 0.f, 64);

  gemm_wmma<128, 32><<<gemm_blocks, T, 0, stream>>>(x, W1, h1, N);
  att_scores<4><<<cdiv((long long)N * 8, T), T, 0, stream>>>(h1, att_src1, att_dst1, a_s, a_d, N);
  edge_max<<<cdiv(Ep, T), T, 0, stream>>>(a_s, a_d, ei, m_ord, E, N);
  edge_exp_sum<<<cdiv(Ep, T), T, 0, stream>>>(a_s, a_d, ei, m_ord, ex, ssum, E, N);
  aggregate<4><<<cdiv((long long)Ep * 8, T), T, 0, stream>>>(h1, ex, ssum, ei, out1, E, N);

  bn_stats<<<cdiv(N, T), T, 0, stream>>>(out1, stats, N);
  bn_apply<<<cdiv((long long)N * 32, T), T, 0, stream>>>(out1, stats, gamma, beta, N);

  // ---- layer 2 (reuse a_s/a_d/m/ssum/ex) ----
  fill_f32<<<cdiv((long long)N * 64, T), T, 0, stream>>>(out2, 0.f, (long long)N * 64);
  fill_f32<<<cdiv((long long)N * 8,  T), T, 0, stream>>>(ssum, 0.f, (long long)N * 8);
  fill_i32<<<cdiv((long long)N * 8,  T), T, 0, stream>>>(m_ord, INT_MIN, (long long)N * 8);

  gemm_wmma<32, 64><<<gemm_blocks, T, 0, stream>>>(out1, W2, h2, N);
  att_scores<8><<<cdiv((long long)N * 8, T), T, 0, stream>>>(h2, att_src2, att_dst2, a_s, a_d, N);
  edge_max<<<cdiv(Ep, T), T, 0, stream>>>(a_s, a_d, ei, m_ord, E, N);
  edge_exp_sum<<<cdiv(Ep, T), T, 0, stream>>>(a_s, a_d, ei, m_ord, ex, ssum, E, N);
  aggregate<8><<<cdiv((long long)Ep * 8, T), T, 0, stream>>>(h2, ex, ssum, ei, out2, E, N);

  finalize<<<cdiv((long long)N * 8, T), T, 0, stream>>>(out2, bias2, out, N);
}